// SVD_10058813407694
// MI455X (gfx1250) — compile-verified
//
#include <hip/hip_runtime.h>
#include <hip/hip_bf16.h>

typedef float v2f __attribute__((ext_vector_type(2)));
typedef float v8f __attribute__((ext_vector_type(8)));

#define KB_EPS 1e-4f

__global__ __launch_bounds__(256)
void kabsch2d_kernel(const float* __restrict__ src,
                     const float* __restrict__ tgt,
                     const float* __restrict__ wgt,
                     float* __restrict__ out,
                     int B, int N) {
    const int b    = blockIdx.x;
    const int tid  = threadIdx.x;
    const int lane = tid & 31;
    const int wid  = tid >> 5;

    const float4* s4 = (const float4*)(src + (size_t)b * N * 2);
    const float4* t4 = (const float4*)(tgt + (size_t)b * N * 2);
    const float2* w2 = (const float2*)(wgt + (size_t)b * N);

    // 9 sufficient statistics:
    // Sw, Ssx, Ssy, Stx, Sty, Cxx=Σw·tx·sx, Cxy=Σw·tx·sy, Cyx=Σw·ty·sx, Cyy=Σw·ty·sy
    float Sw = 0.f, Ssx = 0.f, Ssy = 0.f, Stx = 0.f, Sty = 0.f;
    float Cxx = 0.f, Cxy = 0.f, Cyx = 0.f, Cyy = 0.f;

    const int iters = N >> 9;  // 2 points/thread/iter * 256 threads = 512 points/iter
    for (int k = 0; k < iters; ++k) {
        const int idx = tid + (k << 8);
        float4 s = s4[idx];
        float4 t = t4[idx];
        float2 w = w2[idx];
        if (k + 1 < iters) {
            __builtin_prefetch(&s4[idx + 256], 0, 0);   // global_prefetch_b8
            __builtin_prefetch(&t4[idx + 256], 0, 0);
            __builtin_prefetch(&w2[idx + 256], 0, 0);
        }
        // point 0
        Sw += w.x;
        float wtx = w.x * t.x, wty = w.x * t.y;
        Ssx = fmaf(w.x, s.x, Ssx);  Ssy = fmaf(w.x, s.y, Ssy);
        Stx += wtx;                 Sty += wty;
        Cxx = fmaf(wtx, s.x, Cxx);  Cxy = fmaf(wtx, s.y, Cxy);
        Cyx = fmaf(wty, s.x, Cyx);  Cyy = fmaf(wty, s.y, Cyy);
        // point 1
        Sw += w.y;
        wtx = w.y * t.z;  wty = w.y * t.w;
        Ssx = fmaf(w.y, s.z, Ssx);  Ssy = fmaf(w.y, s.w, Ssy);
        Stx += wtx;                 Sty += wty;
        Cxx = fmaf(wtx, s.z, Cxx);  Cxy = fmaf(wtx, s.w, Cxy);
        Cyx = fmaf(wty, s.z, Cyx);  Cyy = fmaf(wty, s.w, Cyy);
    }

    // ---- wave32 butterfly reduction (9 stats) ----
    float st[9] = {Sw, Ssx, Ssy, Stx, Sty, Cxx, Cxy, Cyx, Cyy};
#pragma unroll
    for (int m = 16; m >= 1; m >>= 1) {
#pragma unroll
        for (int j = 0; j < 9; ++j)
            st[j] += __shfl_xor(st[j], m, 32);
    }

    __shared__ float wp[8][16];  // [wave][stat], stats 9..15 zero
    __shared__ float fin[16];
    if (lane == 0) {
#pragma unroll
        for (int j = 0; j < 9; ++j)  wp[wid][j] = st[j];
#pragma unroll
        for (int j = 9; j < 16; ++j) wp[wid][j] = 0.0f;
    }
    __syncthreads();

    // ---- cross-wave reduction via f32 WMMA: D[m][n] = sum_k A[m][k] with B = ones
    // A layout (ISA 7.12.2, 32-bit A 16x4): lanes 0-15 -> (M=lane, K=0/1),
    // lanes 16-31 -> (M=lane-16, K=2/3). Chain two WMMAs to sum all 8 waves.
    if (wid == 0) {
        const int half = lane >> 4;   // 0 or 1
        const int m    = lane & 15;   // stat row
        v2f a1, a2, bm;
        a1.x = wp[half * 2 + 0][m];
        a1.y = wp[half * 2 + 1][m];
        a2.x = wp[4 + half * 2 + 0][m];
        a2.y = wp[4 + half * 2 + 1][m];
        bm.x = 1.0f; bm.y = 1.0f;
        v8f c = {};
        c = __builtin_amdgcn_wmma_f32_16x16x4_f32(false, a1, false, bm, (short)0, c, false, false);
        c = __builtin_amdgcn_wmma_f32_16x16x4_f32(false, a2, false, bm, (short)0, c, false, false);
        // D layout: VGPR r, lanes 0-15 -> (M=r, N=lane); lanes 16-31 -> (M=8+r)
        if (lane == 0) {
#pragma unroll
            for (int r = 0; r < 8; ++r) fin[r] = c[r];
        }
        if (lane == 16) fin[8] = c[0];
    }
    __syncthreads();

    // ---- closed-form 2D Procrustes (thread 0) ----
    if (tid == 0) {
        const float SwT = fin[0];
        const float we  = SwT + KB_EPS;
        const float inv = 1.0f / we;
        const float sbx = fin[1] * inv, sby = fin[2] * inv;  // src centroid
        const float tbx = fin[3] * inv, tby = fin[4] * inv;  // tgt centroid
        // W_ij = (C_ij - tb_i*Ss_j - sb_j*St_i + tb_i*sb_j*Sw) / we   (i: tgt, j: src)
        const float a  = (fin[5] - tbx * fin[1] - sbx * fin[3] + tbx * sbx * SwT) * inv;
        const float bq = (fin[6] - tbx * fin[2] - sby * fin[3] + tbx * sby * SwT) * inv;
        const float cq = (fin[7] - tby * fin[1] - sbx * fin[4] + tby * sbx * SwT) * inv;
        const float dq = (fin[8] - tby * fin[2] - sby * fin[4] + tby * sby * SwT) * inv;

        const float det2 = a * dq - bq * cq;
        float R00, R01, R10, R11, szz;
        if (det2 >= 0.0f) {
            const float t1 = a + dq, t2 = cq - bq;
            const float r  = sqrtf(t1 * t1 + t2 * t2);
            if (r > 1e-30f) {
                const float ir = 1.0f / r;
                R00 =  t1 * ir; R01 = -t2 * ir;
                R10 =  t2 * ir; R11 =  t1 * ir;
            } else { R00 = 1.f; R01 = 0.f; R10 = 0.f; R11 = 1.f; }
            szz = 1.0f;
        } else {
            const float t3 = a - dq, t4 = bq + cq;
            const float ir = 1.0f / sqrtf(t3 * t3 + t4 * t4);  // det2<0 => r>0
            R00 =  t3 * ir; R01 =  t4 * ir;
            R10 =  t4 * ir; R11 = -t3 * ir;
            szz = -1.0f;
        }

        float* Rout = out + (size_t)b * 9;
        Rout[0] = R00; Rout[1] = R01; Rout[2] = 0.0f;
        Rout[3] = R10; Rout[4] = R11; Rout[5] = 0.0f;
        Rout[6] = 0.0f; Rout[7] = 0.0f; Rout[8] = szz;

        // t_tgt_src_insrc = src_centroid - R^T @ tgt_centroid
        const float vx = sbx - (R00 * tbx + R10 * tby);
        const float vy = sby - (R01 * tbx + R11 * tby);
        // t_src_tgt_intgt = -R @ v
        float* tout = out + (size_t)B * 9 + (size_t)b * 3;
        tout[0] = -(R00 * vx + R01 * vy);
        tout[1] = -(R10 * vx + R11 * vy);
        tout[2] = 0.0f;
    }
}

extern "C" void kernel_launch(void* const* d_in, const int* in_sizes, int n_in,
                              void* d_out, int out_size, void* d_ws, size_t ws_size,
                              hipStream_t stream) {
    (void)n_in; (void)out_size; (void)d_ws; (void)ws_size;
    const float* src = (const float*)d_in[0];
    const float* tgt = (const float*)d_in[1];
    const float* wgt = (const float*)d_in[2];
    float* out = (float*)d_out;
    const int N = 4096;
    const int B = in_sizes[2] / N;  // weights is [B,1,N]
    kabsch2d_kernel<<<B, 256, 0, stream>>>(src, tgt, wgt, out, B, N);
}